// FGD_74448963109375
// MI455X (gfx1250) — compile-verified
//
#include <hip/hip_runtime.h>
#include <math.h>

// ---------------------------------------------------------------------------
// FGD distillation loss, fused for MI455X (gfx1250, wave32, WMMA).
//
// Shapes: B=16, Cs=128, Ct=256, H=W=128 (HW=16384), C2=128.
//
// K1: align GEMM via V_WMMA_F32_16X16X4_F32, fused with:
//       - write s = W@x + b                                (256 MB)
//       - score_s[b,n] = ctx_w . s[:,n], score_t likewise  (softmax inputs)
//       - sumsq partial of (s-t)^2, rowsum partials of (s-t) per (b,c)
// K2: spatial softmax (ctx_b dropped: shift-invariant)
// K3: ctx_{s,t}[b,c] = sum_n {s,t}[b,c,n] * attn_{s,t}[b,n]
// K4: per-sample bottleneck (Conv1x1 -> LN -> ReLU -> Conv1x1) -> tvec
// K5: loss = (sumsq + sum_{b,c} (2 d rowsum + HW d^2)) / (B*Ct*HW),
//     d = tvec_s - tvec_t   (uses s_tr - t_tr = (s-t) + d identity)
// ---------------------------------------------------------------------------

#define B_    16
#define CS    128
#define CT    256
#define HW    16384
#define NPOS  64                  // spatial positions per K1 block
#define NCHUNK (HW / NPOS)        // 256 chunks per batch image
#define NBLK1 (B_ * NCHUNK)       // 4096 K1 blocks
#define KSTEPS (CS / 4)           // 32 wmma K-steps of 4

typedef __attribute__((ext_vector_type(2))) float v2f;
typedef __attribute__((ext_vector_type(8))) float v8f;

// ===========================================================================
// K1: WMMA align-GEMM + fused statistics.
// Block: 256 threads (8 waves). Each block owns (b, 64 spatial positions),
// covers all 256 output channels: wave w handles channel-tiles {2w, 2w+1}
// over 4 sub-tiles of 16 positions. Student K-slab staged in LDS once.
// ===========================================================================
__global__ __launch_bounds__(256) void k1_align_gemm(
    const float* __restrict__ x,      // student (B,CS,HW)
    const float* __restrict__ t,      // teacher (B,CT,HW)
    const float* __restrict__ W,      // align_w (CT,CS)
    const float* __restrict__ bias,   // align_b (CT)
    const float* __restrict__ cw,     // ctx_w (CT)
    float* __restrict__ s_out,        // (B,CT,HW)
    float* __restrict__ scr_s,        // (B,HW)
    float* __restrict__ scr_t,        // (B,HW)
    float* __restrict__ part_rowsum,  // (NBLK1, CT)
    float* __restrict__ sq_part)      // (NBLK1)
{
  __shared__ float xs[CS][NPOS];      // 32 KB student slab [k][n]
  __shared__ float sc_s[8][NPOS];     // per-wave score partials
  __shared__ float sc_t[8][NPOS];
  __shared__ float rsum[CT];          // per-channel delta rowsums (exclusive)
  __shared__ float sqw[8];            // per-wave sum of squares

  const int tid   = threadIdx.x;
  const int blk   = blockIdx.x;
  const int b     = blk >> 8;         // / NCHUNK
  const int chunk = blk & 255;
  const int n0    = chunk * NPOS;

  // zero LDS accumulators
  for (int i = tid; i < 8 * NPOS; i += 256) {
    (&sc_s[0][0])[i] = 0.f;
    (&sc_t[0][0])[i] = 0.f;
  }
  rsum[tid] = 0.f;
  if (tid < 8) sqw[tid] = 0.f;

  // stage student tile: 128 rows x 64 floats, b128 coalesced
  const float* xb = x + (size_t)b * CS * HW + n0;
  for (int i = 0; i < 8; ++i) {
    int id = i * 256 + tid;           // 0..2047 float4 slots
    int k  = id >> 4;
    int q  = id & 15;
    float4 v = *(const float4*)(xb + (size_t)k * HW + q * 4);
    *(float4*)(&xs[k][q * 4]) = v;
  }
  __syncthreads();

  const int wave = tid >> 5;
  const int lane = tid & 31;
  const int half = lane >> 4;         // 0: lanes 0-15, 1: lanes 16-31
  const int lq   = lane & 15;
  const int koff = half * 2;          // A/B fragment K-pair select

  for (int ct2 = 0; ct2 < 2; ++ct2) {
    const int cbase = (wave * 2 + ct2) * 16;
    const int m     = cbase + lq;     // A-matrix row (output channel)

    // Preload A fragments for full K=128 (64 VGPRs); weights are L2-resident.
    v2f areg[KSTEPS];
    const float* wrow = W + (size_t)m * CS;
#pragma unroll
    for (int ks = 0; ks < KSTEPS; ++ks) {
      areg[ks].x = wrow[ks * 4 + koff];
      areg[ks].y = wrow[ks * 4 + koff + 1];
    }
    // Per-lane metadata for D rows half*8+v (C/D VGPR layout).
    float bias8[8], cw8[8];
#pragma unroll
    for (int v = 0; v < 8; ++v) {
      int c = cbase + half * 8 + v;
      bias8[v] = bias[c];
      cw8[v]   = cw[c];
    }

    for (int st = 0; st < 4; ++st) {
      const int nsub = st * 16 + lq;
      v8f acc;
#pragma unroll
      for (int v = 0; v < 8; ++v) acc[v] = bias8[v];   // fold conv bias into C

#pragma unroll
      for (int ks = 0; ks < KSTEPS; ++ks) {
        v2f bf;
        bf.x = xs[ks * 4 + koff][nsub];
        bf.y = xs[ks * 4 + koff + 1][nsub];
        acc = __builtin_amdgcn_wmma_f32_16x16x4_f32(
            false, areg[ks], false, bf, (short)0, acc, false, false);
      }

      // ---- fused epilogue ----
      const int    ng    = n0 + nsub;
      const size_t crow0 = (size_t)(b * CT + cbase + half * 8);
      float ps = 0.f, pt = 0.f, sq = 0.f;
      float dl[8];
#pragma unroll
      for (int v = 0; v < 8; ++v) {
        const float sv = acc[v];
        s_out[(crow0 + v) * HW + ng] = sv;                 // materialize s
        const float tv = t[(crow0 + v) * HW + ng];
        ps += cw8[v] * sv;
        pt += cw8[v] * tv;
        const float d = sv - tv;
        dl[v] = d;
        sq += d * d;
      }
      // combine lane l / l+16 (other 8 channels of same n)
      ps += __shfl_xor(ps, 16);
      pt += __shfl_xor(pt, 16);
      if (half == 0) {                 // exclusive slot per (wave, n)
        sc_s[wave][st * 16 + lq] += ps;
        sc_t[wave][st * 16 + lq] += pt;
      }
      // per-channel rowsum of delta: reduce over 16 lanes of each half
#pragma unroll
      for (int v = 0; v < 8; ++v) {
        float r = dl[v];
        r += __shfl_xor(r, 1);
        r += __shfl_xor(r, 2);
        r += __shfl_xor(r, 4);
        r += __shfl_xor(r, 8);
        if (lq == 0) rsum[cbase + half * 8 + v] += r;  // exclusive channel
      }
      sq += __shfl_xor(sq, 1);
      sq += __shfl_xor(sq, 2);
      sq += __shfl_xor(sq, 4);
      sq += __shfl_xor(sq, 8);
      sq += __shfl_xor(sq, 16);
      if (lane == 0) sqw[wave] += sq;                  // exclusive slot
    }
  }
  __syncthreads();

  // deterministic block-level writebacks (no global atomics anywhere)
  if (tid < NPOS) {
    float ss = 0.f, tt = 0.f;
#pragma unroll
    for (int w = 0; w < 8; ++w) { ss += sc_s[w][tid]; tt += sc_t[w][tid]; }
    scr_s[(size_t)b * HW + n0 + tid] = ss;
    scr_t[(size_t)b * HW + n0 + tid] = tt;
  }
  part_rowsum[(size_t)blk * CT + tid] = rsum[tid];
  if (tid == 0) {
    float s8 = 0.f;
    for (int w = 0; w < 8; ++w) s8 += sqw[w];
    sq_part[blk] = s8;
  }
}

// ===========================================================================
// Reduce rowsum partials: rowsum[b,c] = sum_chunk part[(b*NCHUNK+ch), c]
// ===========================================================================
__global__ __launch_bounds__(256) void k_red_rowsum(
    const float* __restrict__ part, float* __restrict__ rowsum)
{
  const int b = blockIdx.x, c = threadIdx.x;
  float s = 0.f;
  for (int ch = 0; ch < NCHUNK; ++ch)
    s += part[((size_t)(b * NCHUNK + ch)) * CT + c];
  rowsum[b * CT + c] = s;
}

// ===========================================================================
// Spatial softmax, in place. blockIdx: which*? -> b = x>>1, which = x&1.
// scores layout: [which][b][HW] with s first.
// ===========================================================================
__global__ __launch_bounds__(256) void k_softmax(float* __restrict__ scr)
{
  __shared__ float red[256];
  const int b = blockIdx.x >> 1, which = blockIdx.x & 1;
  float* row = scr + (size_t)which * (B_ * HW) + (size_t)b * HW;
  const int tid = threadIdx.x;

  float mx = -INFINITY;
  for (int i = tid; i < HW; i += 256) mx = fmaxf(mx, row[i]);
  red[tid] = mx; __syncthreads();
  for (int s = 128; s > 0; s >>= 1) {
    if (tid < s) red[tid] = fmaxf(red[tid], red[tid + s]);
    __syncthreads();
  }
  mx = red[0]; __syncthreads();

  float sum = 0.f;
  for (int i = tid; i < HW; i += 256) sum += expf(row[i] - mx);
  red[tid] = sum; __syncthreads();
  for (int s = 128; s > 0; s >>= 1) {
    if (tid < s) red[tid] += red[tid + s];
    __syncthreads();
  }
  const float inv = 1.f / red[0];
  __syncthreads();
  for (int i = tid; i < HW; i += 256) row[i] = expf(row[i] - mx) * inv;
}

// ===========================================================================
// Attention pooling: one block per (b,c); streams s and teacher with b128.
// ===========================================================================
__global__ __launch_bounds__(256) void k_ctx(
    const float* __restrict__ s, const float* __restrict__ t,
    const float* __restrict__ attn_s, const float* __restrict__ attn_t,
    float* __restrict__ ctx_s, float* __restrict__ ctx_t)
{
  const int blk = blockIdx.x;
  const int b = blk >> 8, c = blk & 255;
  const float4* sp = (const float4*)(s + ((size_t)b * CT + c) * HW);
  const float4* tp = (const float4*)(t + ((size_t)b * CT + c) * HW);
  const float4* as = (const float4*)(attn_s + (size_t)b * HW);
  const float4* at = (const float4*)(attn_t + (size_t)b * HW);
  const int tid = threadIdx.x;

  float accs = 0.f, acct = 0.f;
  for (int i = tid; i < HW / 4; i += 256) {
    float4 xv = sp[i], av = as[i];
    accs += xv.x * av.x + xv.y * av.y + xv.z * av.z + xv.w * av.w;
    float4 tv = tp[i], bv = at[i];
    acct += tv.x * bv.x + tv.y * bv.y + tv.z * bv.z + tv.w * bv.w;
  }
  __shared__ float rs[256], rt[256];
  rs[tid] = accs; rt[tid] = acct; __syncthreads();
  for (int st = 128; st > 0; st >>= 1) {
    if (tid < st) { rs[tid] += rs[tid + st]; rt[tid] += rt[tid + st]; }
    __syncthreads();
  }
  if (tid == 0) { ctx_s[blk] = rs[0]; ctx_t[blk] = rt[0]; }
}

// ===========================================================================
// Bottleneck transform per (which,b): Conv1x1(256->128), LN(128), ReLU,
// Conv1x1(128->256). ctx/tvec layout: [which][b][CT].
// ===========================================================================
__global__ __launch_bounds__(256) void k_transform(
    const float* __restrict__ ctx,
    const float* __restrict__ w1, const float* __restrict__ b1,
    const float* __restrict__ g,  const float* __restrict__ be,
    const float* __restrict__ w2, const float* __restrict__ b2,
    float* __restrict__ tvec)
{
  const int blk = blockIdx.x;             // 0..31
  const int tid = threadIdx.x;
  __shared__ float cs[CT];
  __shared__ float red[128];
  __shared__ float hrelu[128];

  cs[tid] = ctx[(size_t)blk * CT + tid];
  __syncthreads();

  float hv = 0.f;
  if (tid < 128) {
    hv = b1[tid];
    const float* wr = w1 + (size_t)tid * CT;
    for (int c = 0; c < CT; ++c) hv += wr[c] * cs[c];
  }
  if (tid < 128) red[tid] = hv;
  __syncthreads();
  for (int s = 64; s > 0; s >>= 1) {
    if (tid < s) red[tid] += red[tid + s];
    __syncthreads();
  }
  const float mu = red[0] * (1.f / 128.f);
  __syncthreads();
  if (tid < 128) { float d = hv - mu; red[tid] = d * d; }
  __syncthreads();
  for (int s = 64; s > 0; s >>= 1) {
    if (tid < s) red[tid] += red[tid + s];
    __syncthreads();
  }
  const float rstd = rsqrtf(red[0] * (1.f / 128.f) + 1e-5f);
  __syncthreads();
  if (tid < 128)
    hrelu[tid] = fmaxf((hv - mu) * rstd * g[tid] + be[tid], 0.f);
  __syncthreads();

  float a = b2[tid];
  const float* wr2 = w2 + (size_t)tid * 128;
  for (int j = 0; j < 128; ++j) a += wr2[j] * hrelu[j];
  tvec[(size_t)blk * CT + tid] = a;
}

// ===========================================================================
// Final scalar: loss = (sumsq + sum_{b,c} (2 d rowsum + HW d^2)) / N
// ===========================================================================
__global__ __launch_bounds__(256) void k_final(
    const float* __restrict__ sq_part,
    const float* __restrict__ rowsum,
    const float* __restrict__ tvec,
    float* __restrict__ out)
{
  const int tid = threadIdx.x;
  float acc = 0.f;
  for (int i = tid; i < NBLK1; i += 256) acc += sq_part[i];
  const float* ts = tvec;
  const float* tt = tvec + B_ * CT;
  for (int i = tid; i < B_ * CT; i += 256) {
    float d = ts[i] - tt[i];
    acc += 2.f * d * rowsum[i] + (float)HW * d * d;
  }
  __shared__ float red[256];
  red[tid] = acc; __syncthreads();
  for (int s = 128; s > 0; s >>= 1) {
    if (tid < s) red[tid] += red[tid + s];
    __syncthreads();
  }
  if (tid == 0) out[0] = red[0] / (float)((size_t)B_ * CT * HW);
}

// ===========================================================================
extern "C" void kernel_launch(void* const* d_in, const int* in_sizes, int n_in,
                              void* d_out, int out_size, void* d_ws, size_t ws_size,
                              hipStream_t stream) {
  (void)in_sizes; (void)n_in; (void)out_size; (void)ws_size;
  const float* student = (const float*)d_in[0];
  const float* teacher = (const float*)d_in[1];
  const float* align_w = (const float*)d_in[2];
  const float* align_b = (const float*)d_in[3];
  const float* ctx_w   = (const float*)d_in[4];
  // d_in[5] = ctx_b: drops out (softmax is shift-invariant)
  const float* w1   = (const float*)d_in[6];
  const float* b1   = (const float*)d_in[7];
  const float* ln_g = (const float*)d_in[8];
  const float* ln_b = (const float*)d_in[9];
  const float* w2   = (const float*)d_in[10];
  const float* b2   = (const float*)d_in[11];

  float* ws      = (float*)d_ws;
  float* s_buf   = ws;                                   // B*CT*HW
  float* scores  = s_buf  + (size_t)B_ * CT * HW;        // 2*B*HW (s then t)
  float* part_rs = scores + 2 * (size_t)B_ * HW;         // NBLK1*CT
  float* rowsum  = part_rs + (size_t)NBLK1 * CT;         // B*CT
  float* ctx     = rowsum + B_ * CT;                     // 2*B*CT (s then t)
  float* tvec    = ctx + 2 * B_ * CT;                    // 2*B*CT
  float* sqp     = tvec + 2 * B_ * CT;                   // NBLK1

  float* scr_s = scores;
  float* scr_t = scores + (size_t)B_ * HW;

  k1_align_gemm<<<NBLK1, 256, 0, stream>>>(
      student, teacher, align_w, align_b, ctx_w,
      s_buf, scr_s, scr_t, part_rs, sqp);
  k_red_rowsum<<<B_, CT, 0, stream>>>(part_rs, rowsum);
  k_softmax<<<2 * B_, 256, 0, stream>>>(scores);
  k_ctx<<<B_ * CT, 256, 0, stream>>>(
      s_buf, teacher, scr_s, scr_t, ctx, ctx + B_ * CT);
  k_transform<<<2 * B_, 256, 0, stream>>>(
      ctx, w1, b1, ln_g, ln_b, w2, b2, tvec);
  k_final<<<1, 256, 0, stream>>>(sqp, rowsum, tvec, (float*)d_out);
}